// TrueMambaS6Block_26087631356283
// MI455X (gfx1250) — compile-verified
//
#include <hip/hip_runtime.h>
#include <hip/hip_bf16.h>

#define Bb 4
#define Lt 4096
#define Dm 768
#define Ns 16
#define CH 64                 // chunk length
#define NC (Lt / CH)          // 64 chunks
#define CBDN (NC * Bb * Dm * Ns)
#define BDN  (Bb * Dm * Ns)

typedef float v2f __attribute__((ext_vector_type(2)));
typedef float v8f __attribute__((ext_vector_type(8)));
typedef int   v4i __attribute__((vector_size(4 * sizeof(int))));

// async-to-LDS builtin wants typed v4i pointers: (global src, LDS dst, off, cpol)
#define ASG(p) ((__attribute__((address_space(1))) v4i*)(p))
#define ASL(p) ((__attribute__((address_space(3))) v4i*)(p))

// ---------------------------------------------------------------------------
// delta GEMM: out = softplus(X (16384x768) * W^T (768x768) + bias)
// 256-thread block computes a 128x64 tile. K staged through LDS in 32-wide
// slabs with double-buffered GLOBAL_LOAD_ASYNC_TO_LDS_B128 (ASYNCcnt),
// WMMA operands fed from LDS (padded stride 36 floats -> conflict-free,
// 16B-aligned rows). Each wave: 16-row strip x 4 E-tiles (A-frag reuse x4).
// ---------------------------------------------------------------------------
#define BM 128
#define BE 64
#define KC 32
#define LDST 36               // LDS row stride in floats (144B: 16B aligned)

__global__ __launch_bounds__(256) void gemm_delta_wmma(
    const float* __restrict__ X, const float* __restrict__ W,
    const float* __restrict__ bias, float* __restrict__ out) {
  __shared__ float sX[2][BM * LDST];
  __shared__ float sW[2][BE * LDST];
  const int K = Dm, E = Dm;
  const int tid  = threadIdx.x;
  const int wave = tid >> 5;
  const int lane = tid & 31;
  const int half = lane >> 4, l16 = lane & 15;

  const int tiles_e = E / BE;                 // 12
  const int m0 = (blockIdx.x / tiles_e) * BM;
  const int e0 = (blockIdx.x % tiles_e) * BE;

  auto issue = [&](int buf, int k0) {
    // X slab: 128 rows x 32 floats = 1024 x b128 chunks
#pragma unroll
    for (int i = 0; i < 4; ++i) {
      int q = tid + 256 * i;
      int row = q >> 3, seg = q & 7;
      const float* g = X + (size_t)(m0 + row) * K + k0 + seg * 4;
      float* l = &sX[buf][row * LDST + seg * 4];
      __builtin_amdgcn_global_load_async_to_lds_b128(ASG(g), ASL(l), 0, 0);
    }
    // W slab: 64 rows x 32 floats = 512 x b128 chunks
#pragma unroll
    for (int i = 0; i < 2; ++i) {
      int q = tid + 256 * i;
      int row = q >> 3, seg = q & 7;
      const float* g = W + (size_t)(e0 + row) * K + k0 + seg * 4;
      float* l = &sW[buf][row * LDST + seg * 4];
      __builtin_amdgcn_global_load_async_to_lds_b128(ASG(g), ASL(l), 0, 0);
    }
  };

  v8f acc[4];
#pragma unroll
  for (int j = 0; j < 4; ++j) acc[j] = (v8f){0.f,0.f,0.f,0.f,0.f,0.f,0.f,0.f};

  const int NK = K / KC;                      // 24
  issue(0, 0);
  for (int ko = 0; ko < NK; ++ko) {
    const int buf = ko & 1;
    if (ko + 1 < NK) {
      issue(buf ^ 1, (ko + 1) * KC);
      __builtin_amdgcn_s_wait_asynccnt(6);    // slab ko complete (in-order)
    } else {
      __builtin_amdgcn_s_wait_asynccnt(0);
    }
    __syncthreads();
#pragma unroll
    for (int ks = 0; ks < KC / 4; ++ks) {
      v2f a;
      const float* ap = &sX[buf][(16 * wave + l16) * LDST + 4 * ks + 2 * half];
      a[0] = ap[0]; a[1] = ap[1];
#pragma unroll
      for (int j = 0; j < 4; ++j) {
        v2f b;
        const float* bp = &sW[buf][(16 * j + l16) * LDST + 4 * ks + 2 * half];
        b[0] = bp[0]; b[1] = bp[1];
        acc[j] = __builtin_amdgcn_wmma_f32_16x16x4_f32(
            false, a, false, b, (short)0, acc[j], false, false);
      }
    }
    __syncthreads();
  }

  // epilogue: bias + softplus, C/D layout M = r + 8*half, N = l16
#pragma unroll
  for (int j = 0; j < 4; ++j) {
    int ecol = e0 + 16 * j + l16;
    float bv = bias[ecol];
#pragma unroll
    for (int r = 0; r < 8; ++r) {
      int m = m0 + 16 * wave + r + 8 * half;
      float v = acc[j][r] + bv;
      v = (v > 20.0f) ? v : log1pf(__expf(v));
      out[(size_t)m * E + ecol] = v;
    }
  }
}

// ---------------------------------------------------------------------------
// Small projections (E=16): one wave per 16x16 tile, direct-from-global WMMA.
// ---------------------------------------------------------------------------
__global__ __launch_bounds__(128) void proj_wmma_f32(
    const float* __restrict__ X, const float* __restrict__ W,
    float* __restrict__ out, int M, int K, int E) {
  const int waveId = threadIdx.x >> 5;
  const int lane   = threadIdx.x & 31;
  const int tiles_e = E >> 4;
  const int tile = blockIdx.x * 4 + waveId;
  const int te = tile % tiles_e;
  const int tm = tile / tiles_e;
  const int m0 = tm << 4, e0 = te << 4;
  const int half = lane >> 4, l16 = lane & 15;

  const float* arow = X + (size_t)(m0 + l16) * K + 2 * half;
  const float* brow = W + (size_t)(e0 + l16) * K + 2 * half;

  v8f acc = {0.f, 0.f, 0.f, 0.f, 0.f, 0.f, 0.f, 0.f};
  for (int k = 0; k < K; k += 4) {
    v2f a, b;
    a[0] = arow[k]; a[1] = arow[k + 1];
    b[0] = brow[k]; b[1] = brow[k + 1];
    acc = __builtin_amdgcn_wmma_f32_16x16x4_f32(
        false, a, false, b, (short)0, acc, false, false);
  }
  const int ecol = e0 + l16;
#pragma unroll
  for (int r = 0; r < 8; ++r) {
    int m = m0 + r + 8 * half;
    out[(size_t)m * E + ecol] = acc[r];
  }
}

// ---------------------------------------------------------------------------
// S1: per-(chunk,b,d) local scan from h=0; emits final local state and the
// chunk transition aprod[n] = exp(A_real[d,n] * sum_t delta_t).
// ---------------------------------------------------------------------------
__global__ __launch_bounds__(256) void scan_local(
    const float* __restrict__ x, const float* __restrict__ delta,
    const float* __restrict__ Btp, const float* __restrict__ A_log,
    float* __restrict__ hfin, float* __restrict__ aprod) {
  const int tid = blockIdx.x * blockDim.x + threadIdx.x;
  const int d  = tid % Dm;
  const int cb = tid / Dm;
  const int b  = cb % Bb;
  const int c  = cb / Bb;

  float Ar[Ns], iA[Ns];
#pragma unroll
  for (int n = 0; n < Ns; ++n) {
    float a = -__expf(A_log[d * Ns + n]);
    Ar[n] = a; iA[n] = 1.0f / a;
  }
  float h[Ns];
#pragma unroll
  for (int n = 0; n < Ns; ++n) h[n] = 0.0f;

  float s = 0.0f;
  const int t0 = c * CH;
  const size_t rowoff = ((size_t)b * Lt + t0) * Dm + d;
  const float* dl = delta + rowoff;
  const float* xp = x + rowoff;
  const float* bp = Btp + ((size_t)b * Lt + t0) * Ns;

  for (int t = 0; t < CH; ++t) {
    float dt = dl[(size_t)t * Dm];
    float xt = xp[(size_t)t * Dm];
    s += dt;
#pragma unroll
    for (int n = 0; n < Ns; ++n) {
      float ab = __expf(dt * Ar[n]);
      float bb = (ab - 1.0f) * iA[n] * bp[t * Ns + n];
      h[n] = fmaf(ab, h[n], bb * xt);
    }
  }
  const size_t base = (size_t)tid * Ns;   // [c][b][d][n]
#pragma unroll
  for (int n = 0; n < Ns; ++n) {
    hfin[base + n]  = h[n];
    aprod[base + n] = __expf(Ar[n] * s);
  }
}

// ---------------------------------------------------------------------------
// S2: serial combine across chunks: h0[c] per (b,d,n).
// ---------------------------------------------------------------------------
__global__ __launch_bounds__(256) void scan_combine(
    const float* __restrict__ hfin, const float* __restrict__ aprod,
    float* __restrict__ h0) {
  const int j = blockIdx.x * blockDim.x + threadIdx.x;  // 0 .. B*D*N-1
  float h = 0.0f;
  for (int c = 0; c < NC; ++c) {
    size_t idx = (size_t)c * BDN + j;
    h0[idx] = h;
    h = fmaf(aprod[idx], h, hfin[idx]);
  }
}

// ---------------------------------------------------------------------------
// S3: local scan from true chunk-initial state; emits y. `io` holds delta on
// entry and is overwritten with y (read-before-write within owning thread).
// ---------------------------------------------------------------------------
__global__ __launch_bounds__(256) void scan_final(
    const float* __restrict__ x, const float* __restrict__ Btp,
    const float* __restrict__ Ctp, const float* __restrict__ A_log,
    const float* __restrict__ Dskip, const float* __restrict__ obias,
    const float* __restrict__ h0, float* io) {
  const int tid = blockIdx.x * blockDim.x + threadIdx.x;
  const int d  = tid % Dm;
  const int cb = tid / Dm;
  const int b  = cb % Bb;
  const int c  = cb / Bb;

  float Ar[Ns], iA[Ns];
#pragma unroll
  for (int n = 0; n < Ns; ++n) {
    float a = -__expf(A_log[d * Ns + n]);
    Ar[n] = a; iA[n] = 1.0f / a;
  }
  float h[Ns];
  const size_t base = (size_t)tid * Ns;
#pragma unroll
  for (int n = 0; n < Ns; ++n) h[n] = h0[base + n];

  const int t0 = c * CH;
  const size_t rowoff = ((size_t)b * Lt + t0) * Dm + d;
  float* dl = io + rowoff;                       // delta in, y out
  const float* xp = x + rowoff;
  const float* bp = Btp + ((size_t)b * Lt + t0) * Ns;
  const float* cp = Ctp + ((size_t)b * Lt + t0) * Ns;
  const float Dv = Dskip[d];
  const float ob = obias[d];

  for (int t = 0; t < CH; ++t) {
    float dt = dl[(size_t)t * Dm];
    float xt = xp[(size_t)t * Dm];
    float acc = 0.0f;
#pragma unroll
    for (int n = 0; n < Ns; ++n) {
      float ab = __expf(dt * Ar[n]);
      float bb = (ab - 1.0f) * iA[n] * bp[t * Ns + n];
      h[n] = fmaf(ab, h[n], bb * xt);
      acc = fmaf(cp[t * Ns + n], h[n], acc);
    }
    dl[(size_t)t * Dm] = fmaf(Dv, xt, acc + ob);
  }
}

// ---------------------------------------------------------------------------
extern "C" void kernel_launch(void* const* d_in, const int* in_sizes, int n_in,
                              void* d_out, int out_size, void* d_ws, size_t ws_size,
                              hipStream_t stream) {
  const float* x         = (const float*)d_in[0];
  const float* A_log     = (const float*)d_in[1];
  const float* D_skip    = (const float*)d_in[2];
  const float* out_bias  = (const float*)d_in[3];
  const float* W_delta_w = (const float*)d_in[4];
  const float* W_delta_b = (const float*)d_in[5];
  const float* W_B_w     = (const float*)d_in[6];
  const float* W_C_w     = (const float*)d_in[7];
  float* out = (float*)d_out;

  float* ws    = (float*)d_ws;
  float* Bt    = ws;                          // B*L*N
  float* Ct    = Bt + (size_t)Bb * Lt * Ns;
  float* hfin  = Ct + (size_t)Bb * Lt * Ns;   // NC*B*D*N
  float* aprod = hfin + (size_t)CBDN;
  float* h0    = aprod + (size_t)CBDN;        // total ~40 MB

  const int M = Bb * Lt;  // 16384

  // delta -> d_out (softplus fused), async-LDS double-buffered WMMA GEMM
  gemm_delta_wmma<<<(M / BM) * (Dm / BE), 256, 0, stream>>>(
      x, W_delta_w, W_delta_b, out);
  // B_t, C_t (E=16)
  proj_wmma_f32<<<(M / 16) * (Ns / 16) / 4, 128, 0, stream>>>(
      x, W_B_w, Bt, M, Dm, Ns);
  proj_wmma_f32<<<(M / 16) * (Ns / 16) / 4, 128, 0, stream>>>(
      x, W_C_w, Ct, M, Dm, Ns);

  // chunked scan
  scan_local<<<(NC * Bb * Dm) / 256, 256, 0, stream>>>(x, out, Bt, A_log, hfin, aprod);
  scan_combine<<<BDN / 256, 256, 0, stream>>>(hfin, aprod, h0);
  scan_final<<<(NC * Bb * Dm) / 256, 256, 0, stream>>>(
      x, Bt, Ct, A_log, D_skip, out_bias, h0, out);
}